// GraphSAGEConv_26800595927131
// MI455X (gfx1250) — compile-verified
//
#include <hip/hip_runtime.h>
#include <hip/hip_bf16.h>

typedef __attribute__((ext_vector_type(2))) float v2f;
typedef __attribute__((ext_vector_type(8))) float v8f;

#define D_FEAT 128
#define LN_EPS 1e-5f

// ---------------------------------------------------------------------------
// In-degree per destination node
// ---------------------------------------------------------------------------
__global__ void k_degree(const int* __restrict__ dst, int* __restrict__ deg, int E) {
    int e = blockIdx.x * blockDim.x + threadIdx.x;
    if (e < E) atomicAdd(&deg[dst[e]], 1);
}

__global__ void k_inv(const int* __restrict__ deg, float* __restrict__ inv, int N) {
    int n = blockIdx.x * blockDim.x + threadIdx.x;
    if (n < N) inv[n] = 1.0f / (float)(deg[n] + 1);
}

// ---------------------------------------------------------------------------
// Edge scatter-add: one wave (32 lanes) per edge, float4 per lane (32*4=128)
// feature table (25.6 MB) is L2-resident -> gathers hit L2.
// ---------------------------------------------------------------------------
__global__ __launch_bounds__(256) void k_scatter(const float* __restrict__ feat,
                                                 const int* __restrict__ src,
                                                 const int* __restrict__ dst,
                                                 float* __restrict__ agg, int E) {
    int lane = threadIdx.x & 31;
    int w    = threadIdx.x >> 5;
    long e   = (long)blockIdx.x * 8 + w;
    if (e >= E) return;
    int s = src[e];
    int d = dst[e];
    const float4 v = *(const float4*)(feat + (long)s * D_FEAT + lane * 4);
    float* o = agg + (long)d * D_FEAT + lane * 4;
    atomicAdd(o + 0, v.x);
    atomicAdd(o + 1, v.y);
    atomicAdd(o + 2, v.z);
    atomicAdd(o + 3, v.w);
}

// ---------------------------------------------------------------------------
// Fused: h=(agg+feat)*inv ; h@W^T+b via V_WMMA_F32_16X16X4_F32 ; LayerNorm ; ELU
// Block = 256 threads (8 wave32), 16 nodes per block.
// LDS: Wtp[128*128] (W transposed, adjacent-k pairs interleaved) + hin[16*128]
// ---------------------------------------------------------------------------
__global__ __launch_bounds__(256) void k_fused(const float* __restrict__ feat,
                                               const float* __restrict__ agg,
                                               const float* __restrict__ inv,
                                               const float* __restrict__ Wl,
                                               const float* __restrict__ bl,
                                               const float* __restrict__ gl,
                                               const float* __restrict__ betal,
                                               float* __restrict__ out, int N) {
    extern __shared__ float smem[];
    float* Wtp = smem;                       // 128*128 floats = 64 KB
    float* hin = smem + D_FEAT * D_FEAT;     // 16*128 floats  =  8 KB (reused as hs)

    const int tid = threadIdx.x;
    const long m0 = (long)blockIdx.x * 16;

    // ---- stage W: Wtp[(k>>1)*256 + o*2 + (k&1)] = W[o][k]  (B fragment = one b64 LDS read)
#pragma unroll
    for (int i = 0; i < 16; ++i) {
        int idx4 = tid + i * 256;            // 0..4095 float4s of W
        int o    = idx4 >> 5;                // 0..127 output row
        int k    = (idx4 & 31) << 2;         // 0..124 step 4
        float4 wv = *(const float4*)(Wl + o * D_FEAT + k);
        int p0 = k >> 1;
        Wtp[(p0 + 0) * 256 + o * 2 + 0] = wv.x;
        Wtp[(p0 + 0) * 256 + o * 2 + 1] = wv.y;
        Wtp[(p0 + 1) * 256 + o * 2 + 0] = wv.z;
        Wtp[(p0 + 1) * 256 + o * 2 + 1] = wv.w;
    }
    // ---- stage hin = (agg + feat) * inv for 16 nodes
#pragma unroll
    for (int i = 0; i < 2; ++i) {
        int idx4 = tid + i * 256;            // 0..511 float4s of 16x128 tile
        int r    = idx4 >> 5;
        int c    = (idx4 & 31) << 2;
        long node = m0 + r;
        float s = inv[node];
        float4 a4 = *(const float4*)(agg  + node * D_FEAT + c);
        float4 f4 = *(const float4*)(feat + node * D_FEAT + c);
        float4 h4 = make_float4((a4.x + f4.x) * s, (a4.y + f4.y) * s,
                                (a4.z + f4.z) * s, (a4.w + f4.w) * s);
        *(float4*)(hin + r * D_FEAT + c) = h4;
    }
    __syncthreads();

    const int lane = tid & 31;
    const int wid  = tid >> 5;
    const int col  = lane & 15;
    const int hi   = lane >> 4;              // lane half selects K pair (ISA A/B layout)
    const int o0   = wid * 16;               // this wave's 16 output columns

    v8f acc = {};
    // D(16x16) += A(16x4) * B(4x16), K = 128 -> 32 WMMAs
    for (int k0 = 0; k0 < D_FEAT; k0 += 4) {
        int ka = k0 + (hi << 1);             // lanes 0-15 -> K=k0,k0+1 ; 16-31 -> k0+2,k0+3
        v2f a = *(const v2f*)(hin + col * D_FEAT + ka);
        v2f b = *(const v2f*)(Wtp + (ka >> 1) * 256 + (o0 + col) * 2);
        acc = __builtin_amdgcn_wmma_f32_16x16x4_f32(
            /*neg_a=*/false, a, /*neg_b=*/false, b,
            /*c_mod=*/(short)0, acc, /*reuse_a=*/false, /*reuse_b=*/false);
    }
    __syncthreads();                          // all waves done reading hin

    // ---- spill D tile (+bias) to LDS (reuse hin region as hs[16][128])
    float* hs = hin;
    float bias = bl[o0 + col];
#pragma unroll
    for (int r = 0; r < 8; ++r) {
        int row = r + (hi << 3);              // C/D layout: VGPR r = M=r (lo half) / M=r+8
        hs[row * D_FEAT + o0 + col] = acc[r] + bias;
    }
    __syncthreads();

    // ---- LayerNorm + ELU: each wave handles 2 rows, wave32 shfl reductions
#pragma unroll
    for (int rr = 0; rr < 2; ++rr) {
        int row = wid * 2 + rr;
        long node = m0 + row;
        float v0 = hs[row * D_FEAT + lane +  0];
        float v1 = hs[row * D_FEAT + lane + 32];
        float v2 = hs[row * D_FEAT + lane + 64];
        float v3 = hs[row * D_FEAT + lane + 96];
        float s = v0 + v1 + v2 + v3;
        float q = v0 * v0 + v1 * v1 + v2 * v2 + v3 * v3;
#pragma unroll
        for (int off = 16; off >= 1; off >>= 1) {
            s += __shfl_xor(s, off, 32);
            q += __shfl_xor(q, off, 32);
        }
        float mu   = s * (1.0f / (float)D_FEAT);
        float var  = q * (1.0f / (float)D_FEAT) - mu * mu;
        float rstd = rsqrtf(var + LN_EPS);
        float vv[4] = {v0, v1, v2, v3};
#pragma unroll
        for (int j = 0; j < 4; ++j) {
            int o = lane + j * 32;
            float y = (vv[j] - mu) * rstd * gl[o] + betal[o];
            y = (y > 0.0f) ? y : (__expf(y) - 1.0f);
            out[node * D_FEAT + o] = y;
        }
    }
}

// ---------------------------------------------------------------------------
extern "C" void kernel_launch(void* const* d_in, const int* in_sizes, int n_in,
                              void* d_out, int out_size, void* d_ws, size_t ws_size,
                              hipStream_t stream) {
    const float* features = (const float*)d_in[0];
    const int*   src      = (const int*)d_in[1];
    const int*   dst      = (const int*)d_in[2];
    const float* W        = (const float*)d_in[3];
    const float* b        = (const float*)d_in[4];
    const float* gamma    = (const float*)d_in[5];
    const float* beta     = (const float*)d_in[6];

    const int D     = in_sizes[3] / in_sizes[4];   // 128
    const int ORDER = in_sizes[4] / D;             // 2
    const int N     = in_sizes[0] / D;             // 50000
    const int E     = in_sizes[1];                 // 600000

    // workspace carve (256B aligned slices)
    auto align_up = [](size_t x) { return (x + 255) & ~(size_t)255; };
    char* p = (char*)d_ws;
    int*   deg = (int*)p;   p += align_up((size_t)N * sizeof(int));
    float* inv = (float*)p; p += align_up((size_t)N * sizeof(float));
    float* agg = (float*)p; p += align_up((size_t)N * D * sizeof(float));
    float* fA  = (float*)p; p += align_up((size_t)N * D * sizeof(float));
    float* fB  = (float*)p; p += align_up((size_t)N * D * sizeof(float));

    hipMemsetAsync(deg, 0, (size_t)N * sizeof(int), stream);
    k_degree<<<(E + 255) / 256, 256, 0, stream>>>(dst, deg, E);
    k_inv<<<(N + 255) / 256, 256, 0, stream>>>(deg, inv, N);

    const size_t lds_bytes = (size_t)(D * D + 16 * D) * sizeof(float); // 72 KB

    const float* cur = features;
    for (int i = 0; i < ORDER; ++i) {
        hipMemsetAsync(agg, 0, (size_t)N * D * sizeof(float), stream);
        k_scatter<<<(E + 7) / 8, 256, 0, stream>>>(cur, src, dst, agg, E);
        float* outp = (i == ORDER - 1) ? (float*)d_out : ((i & 1) ? fB : fA);
        k_fused<<<(N + 15) / 16, 256, lds_bytes, stream>>>(
            cur, agg, inv,
            W + (size_t)i * D * D, b + (size_t)i * D,
            gamma + (size_t)i * D, beta + (size_t)i * D,
            outp, N);
        cur = outp;
    }
}